// MultiTaskLoss_70772471103713
// MI455X (gfx1250) — compile-verified
//
#include <hip/hip_runtime.h>
#include <math.h>

typedef __attribute__((ext_vector_type(2))) float v2f;
typedef __attribute__((ext_vector_type(8))) float v8f;

// Wave32 all-lane sum using V_WMMA_F32_16X16X4_F32 with A = ones(16x4).
// D[m][n] = sum_k B[k][n]; putting each lane's value in B-VGPR0 (B-VGPR1 = 0)
// maps the 32 lane values bijectively onto the 2x16 (K,N) slots, so the 16
// column sums partition all 32 values. d[0] at lane L holds colsum[L & 15];
// four shfl_xor steps fold the 16 columns -> every lane holds the wave total.
__device__ __forceinline__ float wave_sum32(float v) {
    v2f a; a[0] = 1.0f; a[1] = 1.0f;     // A-matrix: all ones
    v2f b; b[0] = v;    b[1] = 0.0f;     // B-matrix: lane value in first slot
    v8f c = {};
    v8f d = __builtin_amdgcn_wmma_f32_16x16x4_f32(
        /*neg_a=*/false, a, /*neg_b=*/false, b,
        /*c_mod=*/(short)0, c, /*reuse_a=*/false, /*reuse_b=*/false);
    float s = d[0];
    s += __shfl_xor(s, 1, 32);
    s += __shfl_xor(s, 2, 32);
    s += __shfl_xor(s, 4, 32);
    s += __shfl_xor(s, 8, 32);
    return s;
}

__global__ __launch_bounds__(256)
void mtl_partial_kernel(
    const float* __restrict__ pred_gps,      // [B,2]
    const float* __restrict__ true_lat,      // [B]
    const float* __restrict__ true_lon,      // [B]
    const float* __restrict__ pred_time,     // [B,2]
    const float* __restrict__ hour,          // [B]
    const float* __restrict__ minute,        // [B]
    const float* __restrict__ orient_logits, // [B,8]
    const int*   __restrict__ orientation,   // [B]
    const float* __restrict__ pred_altitude, // [B,1]
    const float* __restrict__ altitude,      // [B]
    const float* __restrict__ gps_scale,     // [2]
    const float* __restrict__ gps_offset,    // [2]
    float* __restrict__ pws,                 // [4, gridDim.x] partial sums
    int B)
{
    const float RAD = 0.017453292519943295f;   // pi/180
    const float sc0 = gps_scale[0],  sc1 = gps_scale[1];
    const float of0 = gps_offset[0], of1 = gps_offset[1];

    float acc_gps = 0.0f, acc_tim = 0.0f, acc_ori = 0.0f, acc_alt = 0.0f;

    const int stride = gridDim.x * blockDim.x;
    for (int i = blockIdx.x * blockDim.x + threadIdx.x; i < B; i += stride) {
        // ---- GPS haversine ----
        float2 pg = ((const float2*)pred_gps)[i];        // global_load_b64
        float plat = fmaf(pg.x, sc0, of0) * RAD;
        float plon = fmaf(pg.y, sc1, of1) * RAD;
        float tlat = fmaf(true_lat[i], sc0, of0) * RAD;
        float tlon = fmaf(true_lon[i], sc1, of1) * RAD;
        float sdlat = sinf(0.5f * (plat - tlat));
        float sdlon = sinf(0.5f * (plon - tlon));
        float a = fmaf(sdlat, sdlat, cosf(tlat) * cosf(plat) * sdlon * sdlon);
        a = fminf(fmaxf(a, 0.0f), 1.0f);
        acc_gps = fmaf(2.0f * 6371.0f, asinf(sqrtf(a)), acc_gps);

        // ---- time MSE ----
        float2 pt = ((const float2*)pred_time)[i];       // global_load_b64
        float dh = pt.x - hour[i];
        float dm = pt.y - minute[i];
        acc_tim += fmaf(dh, dh, dm * dm);

        // ---- orientation NLL (stable log-softmax) ----
        const float4* lp = (const float4*)(orient_logits + (size_t)i * 8);
        float4 l0 = lp[0];                                // 2x global_load_b128
        float4 l1 = lp[1];
        float x0=l0.x, x1=l0.y, x2=l0.z, x3=l0.w;
        float x4=l1.x, x5=l1.y, x6=l1.z, x7=l1.w;
        float m = fmaxf(fmaxf(fmaxf(x0,x1), fmaxf(x2,x3)),
                        fmaxf(fmaxf(x4,x5), fmaxf(x6,x7)));
        float se = expf(x0-m)+expf(x1-m)+expf(x2-m)+expf(x3-m)
                 + expf(x4-m)+expf(x5-m)+expf(x6-m)+expf(x7-m);
        int lab = orientation[i] & 7;
        float xs = x0;                                    // branchless gather
        xs = (lab == 1) ? x1 : xs;
        xs = (lab == 2) ? x2 : xs;
        xs = (lab == 3) ? x3 : xs;
        xs = (lab == 4) ? x4 : xs;
        xs = (lab == 5) ? x5 : xs;
        xs = (lab == 6) ? x6 : xs;
        xs = (lab == 7) ? x7 : xs;
        acc_ori += (m + logf(se)) - xs;

        // ---- altitude MSE ----
        float da = pred_altitude[i] - altitude[i];
        acc_alt = fmaf(da, da, acc_alt);
    }

    // wave reduction via WMMA + shuffles (EXEC all-1s here: full block active)
    acc_gps = wave_sum32(acc_gps);
    acc_tim = wave_sum32(acc_tim);
    acc_ori = wave_sum32(acc_ori);
    acc_alt = wave_sum32(acc_alt);

    __shared__ float sd[4][8];                    // 8 waves per 256-thread block
    const int lane = threadIdx.x & 31;
    const int wv   = threadIdx.x >> 5;
    if (lane == 0) {
        sd[0][wv] = acc_gps; sd[1][wv] = acc_tim;
        sd[2][wv] = acc_ori; sd[3][wv] = acc_alt;
    }
    __syncthreads();
    if (threadIdx.x < 4) {
        float s = 0.0f;
        #pragma unroll
        for (int w = 0; w < 8; ++w) s += sd[threadIdx.x][w];
        pws[(size_t)threadIdx.x * gridDim.x + blockIdx.x] = s;
    }
}

__global__ __launch_bounds__(256)
void mtl_finalize_kernel(const float* __restrict__ pws, float* __restrict__ out,
                         int nb, float invB)
{
    float s0 = 0.0f, s1 = 0.0f, s2 = 0.0f, s3 = 0.0f;
    for (int j = threadIdx.x; j < nb; j += blockDim.x) {   // fixed order: deterministic
        s0 += pws[0 * nb + j];
        s1 += pws[1 * nb + j];
        s2 += pws[2 * nb + j];
        s3 += pws[3 * nb + j];
    }
    s0 = wave_sum32(s0);
    s1 = wave_sum32(s1);
    s2 = wave_sum32(s2);
    s3 = wave_sum32(s3);

    __shared__ float sd[4][8];
    const int lane = threadIdx.x & 31;
    const int wv   = threadIdx.x >> 5;
    if (lane == 0) { sd[0][wv]=s0; sd[1][wv]=s1; sd[2][wv]=s2; sd[3][wv]=s3; }
    __syncthreads();

    if (threadIdx.x == 0) {
        float S0=0.f, S1=0.f, S2=0.f, S3=0.f;
        #pragma unroll
        for (int w = 0; w < 8; ++w) {
            S0 += sd[0][w]; S1 += sd[1][w]; S2 += sd[2][w]; S3 += sd[3][w];
        }
        float gps = S0 * invB;
        float tim = S1 * (0.5f * invB);   // mean over B*2 elements
        float ori = S2 * invB;
        float alt = S3 * invB;
        out[0] = gps + 0.5f * tim + 0.3f * ori + 0.2f * alt;
        out[1] = gps;
        out[2] = tim;
        out[3] = ori;
        out[4] = alt;
    }
}

extern "C" void kernel_launch(void* const* d_in, const int* in_sizes, int n_in,
                              void* d_out, int out_size, void* d_ws, size_t ws_size,
                              hipStream_t stream)
{
    const float* pred_gps      = (const float*)d_in[0];
    const float* true_lat      = (const float*)d_in[1];
    const float* true_lon      = (const float*)d_in[2];
    const float* pred_time     = (const float*)d_in[3];
    const float* hour          = (const float*)d_in[4];
    const float* minute        = (const float*)d_in[5];
    const float* orient_logits = (const float*)d_in[6];
    const int*   orientation   = (const int*)  d_in[7];
    const float* pred_altitude = (const float*)d_in[8];
    const float* altitude      = (const float*)d_in[9];
    const float* gps_scale     = (const float*)d_in[10];
    const float* gps_offset    = (const float*)d_in[11];

    const int B = in_sizes[1];            // true_lat has B elements

    const int T = 256;
    int NB = 2048;                        // needs 4*NB*4 = 32 KB of d_ws
    size_t need = (size_t)4 * NB * sizeof(float);
    if (need > ws_size) {
        NB = (int)(ws_size / (4 * sizeof(float)));
        if (NB < 1) NB = 1;
    }

    float* pws = (float*)d_ws;
    float* out = (float*)d_out;

    mtl_partial_kernel<<<NB, T, 0, stream>>>(
        pred_gps, true_lat, true_lon, pred_time, hour, minute,
        orient_logits, orientation, pred_altitude, altitude,
        gps_scale, gps_offset, pws, B);

    mtl_finalize_kernel<<<1, T, 0, stream>>>(pws, out, NB, 1.0f / (float)B);
}